// CrossAttention_63221918597767
// MI455X (gfx1250) — compile-verified
//
#include <hip/hip_runtime.h>
#include <math.h>

// ---------------------------------------------------------------------------
// CrossAttention for MI455X (gfx1250): f32 WMMA 16x16x4 for all GEMM-shaped
// stages, double-buffered async global->LDS (ASYNCcnt) tile staging via
// inline CDNA5 asm, plain VALU kernels for depthwise convs and row norms.
// ---------------------------------------------------------------------------

typedef float v2f __attribute__((ext_vector_type(2)));
typedef float v8f __attribute__((ext_vector_type(8)));

#define HW    16384   // 128*128
#define CDIM  192
#define HEADS 4
#define CH    48      // CDIM / HEADS
#define BATCH 8

__device__ __forceinline__ v8f wmma_f32(v2f a, v2f b, v8f c) {
  // 8-arg pattern: (neg_a, A, neg_b, B, c_mod, C, reuse_a, reuse_b)
  return __builtin_amdgcn_wmma_f32_16x16x4_f32(false, a, false, b, (short)0, c,
                                               false, false);
}

// Async 16-byte global -> LDS copy (one lane moves 16B). Tracked by ASYNCcnt.
// Both addresses must be 16B aligned. GV mode: vdst = LDS byte offset VGPR,
// vaddr = 64-bit global address pair.
__device__ __forceinline__ void async_b128(void* lds_dst, const void* g_src) {
  unsigned l = (unsigned)(size_t)lds_dst;            // low 32 bits = LDS offset
  unsigned long long g = (unsigned long long)(size_t)g_src;
  asm volatile("global_load_async_to_lds_b128 %0, %1, off"
               :: "v"(l), "v"(g) : "memory");
}

// Wait for this wave's async copies, then make LDS visible workgroup-wide.
__device__ __forceinline__ void async_wait_and_sync() {
  asm volatile("s_wait_asynccnt 0" ::: "memory");
  __syncthreads();
}

// ---------------------------------------------------------------------------
// 1x1 conv as per-batch GEMM: Out[b] (Cout x HW) = W (Cout x Cin) @ X[b] (Cin x HW)
// Block tile 64(M) x 128(N); 8 waves: wave w -> M-tile (w>>1), N-half
// (w&1)*64. Each wave owns a 16x64 strip = 4 accumulators, reusing one A
// fragment across 4 WMMAs. K chunks of 16 staged through double-buffered LDS:
// async loads for chunk i+1 overlap WMMA compute of chunk i.
// ---------------------------------------------------------------------------
__global__ __launch_bounds__(256) void gemm1x1_kernel(
    const float* __restrict__ W, const float* __restrict__ X,
    float* __restrict__ Out, int Cout, int Cin) {
  __shared__ float Wl[2][64][20];   // 80B rows: 16B aligned, bank-safe columns
  __shared__ float Xl[2][16][132];  // 528B rows: 16B aligned

  const int tid  = threadIdx.x;
  const int lane = tid & 31;
  const int wave = tid >> 5;
  const int mt   = wave >> 1;     // 0..3  (16-row M tile)
  const int nq   = wave & 1;      // 0..1  (64-col N half)
  const int bx   = blockIdx.x;    // N tile of 128
  const int by   = blockIdx.y;    // M tile of 64
  const int b    = blockIdx.z;

  const float* Xb = X + (size_t)b * Cin * HW;
  float*       Ob = Out + (size_t)b * Cout * HW;

  const int mloc = lane & 15;
  const int ksel = (lane >> 4) << 1;   // 0 or 2

  auto load_tiles = [&](int pb, int k0) {
    // W tile 64x16: one b128 per thread
    int r = tid >> 2, c4 = (tid & 3) * 4;
    async_b128(&Wl[pb][r][c4], &W[(size_t)(by * 64 + r) * Cin + k0 + c4]);
    // X tile 16x128: two b128 per thread
    for (int i = tid; i < 512; i += 256) {
      int rr = i >> 5, cc4 = (i & 31) * 4;
      async_b128(&Xl[pb][rr][cc4], &Xb[(size_t)(k0 + rr) * HW + bx * 128 + cc4]);
    }
  };

  v8f acc[4] = {{}, {}, {}, {}};

  load_tiles(0, 0);
  async_wait_and_sync();

  int p = 0;
  for (int k0 = 0; k0 < Cin; k0 += 16) {
    if (k0 + 16 < Cin) load_tiles(p ^ 1, k0 + 16);   // prefetch next chunk

#pragma unroll
    for (int kk = 0; kk < 16; kk += 4) {
      v2f a;
      a.x = Wl[p][mt * 16 + mloc][kk + ksel];
      a.y = Wl[p][mt * 16 + mloc][kk + ksel + 1];
#pragma unroll
      for (int t = 0; t < 4; ++t) {
        v2f bf;
        bf.x = Xl[p][kk + ksel][nq * 64 + t * 16 + mloc];
        bf.y = Xl[p][kk + ksel + 1][nq * 64 + t * 16 + mloc];
        acc[t] = wmma_f32(a, bf, acc[t]);
      }
    }
    async_wait_and_sync();
    p ^= 1;
  }

#pragma unroll
  for (int t = 0; t < 4; ++t) {
#pragma unroll
    for (int j = 0; j < 8; ++j) {
      int row = by * 64 + mt * 16 + j + ((lane >> 4) << 3);
      int col = bx * 128 + nq * 64 + t * 16 + mloc;
      Ob[(size_t)row * HW + col] = acc[t][j];
    }
  }
}

// ---------------------------------------------------------------------------
// Depthwise KxK conv, pad = K/2, on (B, C, 128, 128). One thread per output.
// ---------------------------------------------------------------------------
__global__ __launch_bounds__(256) void dwconv_kernel(
    const float* __restrict__ in, const float* __restrict__ w,
    float* __restrict__ out, int C, int K, int pad, int total) {
  int idx = blockIdx.x * 256 + threadIdx.x;
  if (idx >= total) return;
  int x  = idx & 127;
  int y  = (idx >> 7) & 127;
  int bc = idx >> 14;          // b*C + c
  int c  = bc % C;

  const float* ip = in + (size_t)bc * HW;
  const float* wp = w + (size_t)c * K * K;

  float s = 0.f;
  for (int dy = 0; dy < K; ++dy) {
    int yy = y + dy - pad;
    if (yy < 0 || yy >= 128) continue;
    for (int dx = 0; dx < K; ++dx) {
      int xx = x + dx - pad;
      if (xx < 0 || xx >= 128) continue;
      s += ip[yy * 128 + xx] * wp[dy * K + dx];
    }
  }
  out[(size_t)idx] = s;
}

// ---------------------------------------------------------------------------
// Per-row inverse L2 norm over the spatial dim (16384, contiguous).
// Row r -> batch r/CDIM, channel r%CDIM inside a tensor with chan_total
// channels per batch at chan_offset (lets us norm k inside the kv buffer).
// ---------------------------------------------------------------------------
__global__ __launch_bounds__(256) void rownorm_kernel(
    const float* __restrict__ src, float* __restrict__ inv,
    int chan_total, int chan_offset) {
  __shared__ float red[256];
  int r = blockIdx.x;
  int b = r / CDIM, c = r % CDIM;
  const float* p = src + ((size_t)b * chan_total + chan_offset + c) * HW;

  float s = 0.f;
  for (int i = threadIdx.x; i < HW; i += 256) {
    float v = p[i];
    s += v * v;
  }
  red[threadIdx.x] = s;
  __syncthreads();
  for (int off = 128; off > 0; off >>= 1) {
    if (threadIdx.x < off) red[threadIdx.x] += red[threadIdx.x + off];
    __syncthreads();
  }
  if (threadIdx.x == 0) inv[r] = 1.0f / fmaxf(sqrtf(red[0]), 1e-12f);
}

// ---------------------------------------------------------------------------
// attn = softmax( (q/||q||) @ (k/||k||)^T * temperature ) per (b, head).
// One block per (b,head): 9 waves, each owns one 16x16 tile of the 48x48
// score matrix; K=16384 streamed through double-buffered LDS in 64-wide
// chunks (256 iterations): async loads of chunk i+1 overlap WMMA of chunk i.
// Norm scaling is folded in post-accumulation: S * invq[c] * invk[d] * T.
// ---------------------------------------------------------------------------
__global__ __launch_bounds__(288) void attn_kernel(
    const float* __restrict__ q, const float* __restrict__ kv,
    const float* __restrict__ invq, const float* __restrict__ invk,
    const float* __restrict__ temp, float* __restrict__ attn) {
  __shared__ float qs[2][CH][68];   // 272B rows: 16B aligned
  __shared__ float ks[2][CH][68];
  __shared__ float S[CH][CH];

  const int tid  = threadIdx.x;
  const int lane = tid & 31;
  const int wave = tid >> 5;          // 0..8
  const int mt   = wave / 3;
  const int nt   = wave % 3;
  const int bh   = blockIdx.x;
  const int b    = bh >> 2;
  const int head = bh & 3;
  const int mloc = lane & 15;
  const int ksel = (lane >> 4) << 1;

  auto load_chunk = [&](int pb, int n0) {
    // q/k chunks: 48x64 floats each = 768 b128 transfers per matrix
    for (int i = tid; i < 768; i += 288) {
      int r = i >> 4, c4 = (i & 15) * 4;
      int ch = head * CH + r;
      async_b128(&qs[pb][r][c4], &q[((size_t)b * CDIM + ch) * HW + n0 + c4]);
      async_b128(&ks[pb][r][c4], &kv[((size_t)b * 2 * CDIM + ch) * HW + n0 + c4]);
    }
  };

  v8f acc = {};

  load_chunk(0, 0);
  async_wait_and_sync();

  int p = 0;
  for (int n0 = 0; n0 < HW; n0 += 64) {
    if (n0 + 64 < HW) load_chunk(p ^ 1, n0 + 64);    // prefetch next chunk

#pragma unroll
    for (int kk = 0; kk < 64; kk += 4) {
      v2f a, bb;
      a.x  = qs[p][mt * 16 + mloc][kk + ksel];
      a.y  = qs[p][mt * 16 + mloc][kk + ksel + 1];
      bb.x = ks[p][nt * 16 + mloc][kk + ksel];   // B[k][d] = k[d][k] (q @ k^T)
      bb.y = ks[p][nt * 16 + mloc][kk + ksel + 1];
      acc = wmma_f32(a, bb, acc);
    }
    async_wait_and_sync();
    p ^= 1;
  }

#pragma unroll
  for (int j = 0; j < 8; ++j)
    S[mt * 16 + j + ((lane >> 4) << 3)][nt * 16 + mloc] = acc[j];
  __syncthreads();

  if (tid < CH) {
    const int row = tid;
    const float iq = invq[b * CDIM + head * CH + row];
    const float t  = temp[head];
    float buf[CH];
    float mx = -1e30f;
    for (int d = 0; d < CH; ++d) {
      float v = S[row][d] * iq * invk[b * CDIM + head * CH + d] * t;
      buf[d] = v;
      mx = fmaxf(mx, v);
    }
    float sum = 0.f;
    for (int d = 0; d < CH; ++d) {
      buf[d] = __expf(buf[d] - mx);
      sum += buf[d];
    }
    float is = 1.0f / sum;
    for (int d = 0; d < CH; ++d)
      attn[((size_t)bh * CH + row) * CH + d] = buf[d] * is;
  }
}

// ---------------------------------------------------------------------------
// out = attn (48x48) @ v (48x16384) per (b, head).
// Grid: (HW/64 n-chunks, 32 bh). 12 waves: wave -> (mt = w/4, nt = w%4).
// Single-shot async-staged tiles (one chunk per block, nothing to overlap).
// ---------------------------------------------------------------------------
__global__ __launch_bounds__(384) void attnout_kernel(
    const float* __restrict__ attn, const float* __restrict__ kv,
    float* __restrict__ out) {
  __shared__ float As[CH][52];    // 208B rows: 16B aligned, bank-safe columns
  __shared__ float Vs[CH][68];

  const int tid  = threadIdx.x;
  const int lane = tid & 31;
  const int wave = tid >> 5;      // 0..11
  const int mt   = wave >> 2;
  const int nt   = wave & 3;
  const int bh   = blockIdx.y;
  const int b    = bh >> 2;
  const int head = bh & 3;
  const int n0   = blockIdx.x * 64;

  // attn 48x48 = 576 b128; V chunk 48x64 = 768 b128
  for (int i = tid; i < 576; i += 384) {
    int r = i / 12, c4 = (i % 12) * 4;
    async_b128(&As[r][c4], &attn[((size_t)bh * CH + r) * CH + c4]);
  }
  for (int i = tid; i < 768; i += 384) {
    int r = i >> 4, c4 = (i & 15) * 4;
    async_b128(&Vs[r][c4],
               &kv[((size_t)b * 2 * CDIM + CDIM + head * CH + r) * HW + n0 + c4]);
  }
  async_wait_and_sync();

  const int mloc = lane & 15;
  const int ksel = (lane >> 4) << 1;
  v8f acc = {};
#pragma unroll
  for (int kk = 0; kk < CH; kk += 4) {
    v2f a, bb;
    a.x  = As[mt * 16 + mloc][kk + ksel];
    a.y  = As[mt * 16 + mloc][kk + ksel + 1];
    bb.x = Vs[kk + ksel][nt * 16 + mloc];
    bb.y = Vs[kk + ksel + 1][nt * 16 + mloc];
    acc = wmma_f32(a, bb, acc);
  }

#pragma unroll
  for (int j = 0; j < 8; ++j) {
    int row = mt * 16 + j + ((lane >> 4) << 3);
    int col = nt * 16 + mloc;
    out[((size_t)b * CDIM + head * CH + row) * HW + n0 + col] = acc[j];
  }
}

// ---------------------------------------------------------------------------
extern "C" void kernel_launch(void* const* d_in, const int* in_sizes, int n_in,
                              void* d_out, int out_size, void* d_ws, size_t ws_size,
                              hipStream_t stream) {
  const float* x           = (const float*)d_in[0];
  const float* y           = (const float*)d_in[1];
  const float* q_w         = (const float*)d_in[2];
  const float* q_dw_w      = (const float*)d_in[3];
  const float* kv_w        = (const float*)d_in[4];
  const float* kv_dw_w     = (const float*)d_in[5];
  const float* proj_w      = (const float*)d_in[6];
  const float* temperature = (const float*)d_in[7];
  float* out = (float*)d_out;

  char* ws   = (char*)d_ws;
  size_t off = 0;
  auto alloc = [&](size_t bytes) -> void* {
    void* p = ws + off;
    off += (bytes + 255) & ~(size_t)255;
    return p;
  };

  const size_t planeB = (size_t)BATCH * CDIM * HW * sizeof(float);
  float* qpre  = (float*)alloc(planeB);
  float* qbuf  = (float*)alloc(planeB);
  float* kvpre = (float*)alloc(2 * planeB);
  float* kvbuf = (float*)alloc(2 * planeB);
  float* invq  = (float*)alloc((size_t)BATCH * CDIM * sizeof(float));
  float* invk  = (float*)alloc((size_t)BATCH * CDIM * sizeof(float));
  float* attnb = (float*)alloc((size_t)BATCH * HEADS * CH * CH * sizeof(float));
  float* obuf  = qpre;  // qpre dead after its depthwise conv; reuse for attn out

  // q = dw3x3( q_w @ x ), kv = dw7x7( kv_w @ y )
  gemm1x1_kernel<<<dim3(HW / 128, CDIM / 64, BATCH), 256, 0, stream>>>(
      q_w, x, qpre, CDIM, CDIM);
  gemm1x1_kernel<<<dim3(HW / 128, (2 * CDIM) / 64, BATCH), 256, 0, stream>>>(
      kv_w, y, kvpre, 2 * CDIM, CDIM);

  int totq = BATCH * CDIM * HW;
  dwconv_kernel<<<(totq + 255) / 256, 256, 0, stream>>>(
      qpre, q_dw_w, qbuf, CDIM, 3, 1, totq);
  int totkv = BATCH * 2 * CDIM * HW;
  dwconv_kernel<<<(totkv + 255) / 256, 256, 0, stream>>>(
      kvpre, kv_dw_w, kvbuf, 2 * CDIM, 7, 3, totkv);

  rownorm_kernel<<<BATCH * CDIM, 256, 0, stream>>>(qbuf, invq, CDIM, 0);
  rownorm_kernel<<<BATCH * CDIM, 256, 0, stream>>>(kvbuf, invk, 2 * CDIM, 0);

  attn_kernel<<<BATCH * HEADS, 288, 0, stream>>>(
      qbuf, kvbuf, invq, invk, temperature, attnb);

  attnout_kernel<<<dim3(HW / 64, BATCH * HEADS), 384, 0, stream>>>(
      attnb, kvbuf, obuf);

  gemm1x1_kernel<<<dim3(HW / 128, CDIM / 64, BATCH), 256, 0, stream>>>(
      proj_w, obuf, out, CDIM, CDIM);
}